// Parallel_decoder_41257455845609
// MI455X (gfx1250) — compile-verified
//
#include <hip/hip_runtime.h>
#include <hip/hip_bf16.h>
#include <math.h>

// ---------------------------------------------------------------------------
// CDNA5 (gfx1250) implementation. All heavy GEMMs run on the matrix pipe via
// V_WMMA_F32_16X16X4_F32 (fp32 in/out — this network is gate/softmax heavy).
// wave32; blocks are 4 waves (128 threads) for the WMMA kernels.
// ---------------------------------------------------------------------------

typedef __attribute__((ext_vector_type(2))) float v2f;
typedef __attribute__((ext_vector_type(8))) float v8f;

static __device__ __forceinline__ v8f wmma4(v2f a, v2f b, v8f c) {
  // (neg_a, A, neg_b, B, c_mod, C, reuse_a, reuse_b)
  return __builtin_amdgcn_wmma_f32_16x16x4_f32(false, a, false, b, (short)0, c,
                                               false, false);
}

// A fragment: 16x4 tile, row-major source. Lanes 0-15 hold K={k0,k0+1},
// lanes 16-31 hold K={k0+2,k0+3}; lane&15 selects the M row.
static __device__ __forceinline__ v2f fragA(const float* p, int ld, int row0,
                                            int k0, int lane) {
  const int hl = lane >> 4, r = lane & 15;
  v2f a;
  a.x = p[(row0 + r) * ld + k0 + 2 * hl];
  a.y = p[(row0 + r) * ld + k0 + 2 * hl + 1];
  return a;
}

// B fragment: 4x16 tile, row-major source (K rows, N cols).
static __device__ __forceinline__ v2f fragB(const float* p, int ld, int k0,
                                            int col0, int lane) {
  const int hl = lane >> 4, r = lane & 15;
  v2f b;
  b.x = p[(k0 + 2 * hl) * ld + col0 + r];
  b.y = p[(k0 + 2 * hl + 1) * ld + col0 + r];
  return b;
}

// ---------------------------------------------------------------------------
// Kernel 1: A = softmax(relu(E @ E^T), axis=1).  E: (512,16), A: (512,512)
// One block per row, 256 threads, each thread owns 2 columns.
// ---------------------------------------------------------------------------
__global__ __launch_bounds__(256) void k_adj(const float* __restrict__ E,
                                             float* __restrict__ A) {
  const int n = blockIdx.x, tid = threadIdx.x;
  __shared__ float en[16];
  __shared__ float red[256];
  if (tid < 16) en[tid] = E[n * 16 + tid];
  __syncthreads();
  float s0 = 0.f, s1 = 0.f;
#pragma unroll
  for (int e = 0; e < 16; ++e) {
    s0 += en[e] * E[tid * 16 + e];
    s1 += en[e] * E[(tid + 256) * 16 + e];
  }
  const float v0 = fmaxf(s0, 0.f), v1 = fmaxf(s1, 0.f);
  red[tid] = fmaxf(v0, v1);
  __syncthreads();
  for (int sft = 128; sft > 0; sft >>= 1) {
    if (tid < sft) red[tid] = fmaxf(red[tid], red[tid + sft]);
    __syncthreads();
  }
  const float mx = red[0];
  __syncthreads();
  const float e0 = __expf(v0 - mx), e1 = __expf(v1 - mx);
  red[tid] = e0 + e1;
  __syncthreads();
  for (int sft = 128; sft > 0; sft >>= 1) {
    if (tid < sft) red[tid] += red[tid + sft];
    __syncthreads();
  }
  const float inv = 1.f / red[0];
  A[n * 512 + tid] = e0 * inv;
  A[n * 512 + tid + 256] = e1 * inv;
}

// ---------------------------------------------------------------------------
// Kernel 2: S2 = 2*A@A - I  via WMMA.  Block: 16-row panel of A in LDS,
// 4 waves each own one 16-col tile; K=512.
// ---------------------------------------------------------------------------
__global__ __launch_bounds__(128) void k_cheb(const float* __restrict__ A,
                                              float* __restrict__ S2) {
  const int tid = threadIdx.x, lane = tid & 31, wid = tid >> 5;
  const int m0 = blockIdx.x * 16;
  __shared__ float arow[16][512];  // 32 KB panel
  for (int idx = tid; idx < 16 * 512; idx += 128)
    arow[idx >> 9][idx & 511] = A[(m0 + (idx >> 9)) * 512 + (idx & 511)];
  __syncthreads();
  const int n0 = (blockIdx.y * 4 + wid) * 16;
  v8f c = {0.f, 0.f, 0.f, 0.f, 0.f, 0.f, 0.f, 0.f};
#pragma unroll 4
  for (int kk = 0; kk < 512; kk += 4) {
    v2f a = fragA(&arow[0][0], 512, 0, kk, lane);
    v2f b = fragB(A, 512, kk, n0, lane);
    c = wmma4(a, b, c);
  }
  const int hl = lane >> 4, r = lane & 15;
#pragma unroll
  for (int rv = 0; rv < 8; ++rv) {
    const int row = m0 + rv + 8 * hl, col = n0 + r;
    S2[row * 512 + col] = 2.f * c[rv] - (row == col ? 1.f : 0.f);
  }
}

// ---------------------------------------------------------------------------
// Kernel 3: temporal attention -> de_input (768,512,64).
// Block = (b, 32-row n-tile), 4 waves.  s==1 so attn is a softmax over t of
// per-head q.k dots; out = aw * v.  de = X@W0 + out@W1 + b (X@W0 hoisted).
// ---------------------------------------------------------------------------
__global__ __launch_bounds__(128) void k_attn(
    const float* __restrict__ hn, const float* __restrict__ ne1,
    const float* __restrict__ ne2, const float* __restrict__ Wq,
    const float* __restrict__ bq, const float* __restrict__ Wk,
    const float* __restrict__ bk, const float* __restrict__ Wv,
    const float* __restrict__ bv, const float* __restrict__ taw,
    const float* __restrict__ tab, float* __restrict__ de) {
  const int tid = threadIdx.x, lane = tid & 31, wid = tid >> 5;
  const int n0 = blockIdx.x * 32, b = blockIdx.y;
  __shared__ float aBuf[32][96];     // [STE | X] concat input
  __shared__ float kB[32][64];
  __shared__ float vB[32][64];
  __shared__ float qB[32][64];       // q, later reused for aw*v
  __shared__ float lg[12][32][8];    // logits -> attention weights

  // Build KV input: cols 0..31 = STE_P (ne1[:, -1, :]), cols 32..95 = X rows.
  for (int idx = tid; idx < 32 * 96; idx += 128) {
    const int rr = idx / 96, cc = idx % 96;
    aBuf[rr][cc] = (cc < 32) ? ne1[(b * 12 + 11) * 32 + cc]
                             : hn[(b * 512 + n0 + rr) * 64 + (cc - 32)];
  }
  __syncthreads();

  // k = relu(kv@Wk+bk), v = relu(kv@Wv+bv): 16 tile jobs over 4 waves.
  for (int j = wid; j < 16; j += 4) {
    const int mat = j >> 3, mt = (j >> 2) & 1, ot = j & 3;
    const float* W = mat ? Wv : Wk;
    const float* bias = mat ? bv : bk;
    v8f c = {0.f, 0.f, 0.f, 0.f, 0.f, 0.f, 0.f, 0.f};
#pragma unroll 4
    for (int kk = 0; kk < 96; kk += 4) {
      v2f a = fragA(&aBuf[0][0], 96, 16 * mt, kk, lane);
      v2f bb = fragB(W, 64, kk, 16 * ot, lane);
      c = wmma4(a, bb, c);
    }
    const int hl = lane >> 4, r = lane & 15;
    const float bsv = bias[16 * ot + r];
    float(*dst)[64] = mat ? vB : kB;
#pragma unroll
    for (int rv = 0; rv < 8; ++rv)
      dst[16 * mt + rv + 8 * hl][16 * ot + r] = fmaxf(c[rv] + bsv, 0.f);
  }
  __syncthreads();

  // D0 = X_tile @ ta_w[0] (t-independent part of the combine einsum).
  v8f accD0[2];
#pragma unroll
  for (int jj = 0; jj < 2; ++jj) {
    v8f c = {0.f, 0.f, 0.f, 0.f, 0.f, 0.f, 0.f, 0.f};
#pragma unroll 4
    for (int kk = 0; kk < 64; kk += 4) {
      v2f a = fragA(&aBuf[0][0], 96, 16 * jj, 32 + kk, lane);  // X columns
      v2f bb = fragB(taw, 64, kk, 16 * wid, lane);
      c = wmma4(a, bb, c);
    }
    accD0[jj] = c;
  }

  // q per horizon step + per-head logits.
  for (int t = 0; t < 12; ++t) {
    __syncthreads();
    for (int idx = tid; idx < 32 * 32; idx += 128)
      aBuf[idx >> 5][idx & 31] = ne2[(b * 12 + t) * 32 + (idx & 31)];
    __syncthreads();
#pragma unroll
    for (int jj = 0; jj < 2; ++jj) {
      v8f c = {0.f, 0.f, 0.f, 0.f, 0.f, 0.f, 0.f, 0.f};
#pragma unroll 4
      for (int kk = 0; kk < 96; kk += 4) {
        v2f a = fragA(&aBuf[0][0], 96, 16 * jj, kk, lane);
        v2f bb = fragB(Wq, 64, kk, 16 * wid, lane);
        c = wmma4(a, bb, c);
      }
      const int hl = lane >> 4, r = lane & 15;
      const float bsv = bq[16 * wid + r];
#pragma unroll
      for (int rv = 0; rv < 8; ++rv)
        qB[16 * jj + rv + 8 * hl][16 * wid + r] = fmaxf(c[rv] + bsv, 0.f);
    }
    __syncthreads();
    for (int p = tid; p < 256; p += 128) {
      const int nn = p >> 3, d = p & 7;
      float s = 0.f;
#pragma unroll
      for (int cch = 0; cch < 8; ++cch)
        s += qB[nn][8 * d + cch] * kB[nn][8 * d + cch];
      lg[t][nn][d] = s;
    }
  }
  __syncthreads();

  // Softmax over t per (n, head).
  for (int p = tid; p < 256; p += 128) {
    const int nn = p >> 3, d = p & 7;
    float mx = -1e30f;
    for (int t = 0; t < 12; ++t) mx = fmaxf(mx, lg[t][nn][d]);
    float sum = 0.f;
    for (int t = 0; t < 12; ++t) {
      const float e = __expf(lg[t][nn][d] - mx);
      lg[t][nn][d] = e;
      sum += e;
    }
    const float inv = 1.f / sum;
    for (int t = 0; t < 12; ++t) lg[t][nn][d] *= inv;
  }
  __syncthreads();

  // de[t] = D0 + (aw[t] * v) @ ta_w[1] + ta_b
  for (int t = 0; t < 12; ++t) {
    for (int idx = tid; idx < 32 * 64; idx += 128) {
      const int rr = idx >> 6, cc = idx & 63;
      qB[rr][cc] = lg[t][rr][cc >> 3] * vB[rr][cc];
    }
    __syncthreads();
#pragma unroll
    for (int jj = 0; jj < 2; ++jj) {
      v8f c = accD0[jj];
#pragma unroll 4
      for (int kk = 0; kk < 64; kk += 4) {
        v2f a = fragA(&qB[0][0], 64, 16 * jj, kk, lane);
        v2f bb = fragB(taw + 4096, 64, kk, 16 * wid, lane);
        c = wmma4(a, bb, c);
      }
      const int hl = lane >> 4, r = lane & 15;
      const float bsv = tab[16 * wid + r];
#pragma unroll
      for (int rv = 0; rv < 8; ++rv) {
        const int nrow = n0 + 16 * jj + rv + 8 * hl;
        de[((b * 12 + t) * 512 + nrow) * 64 + 16 * wid + r] = c[rv] + bsv;
      }
    }
    __syncthreads();
  }
}

// ---------------------------------------------------------------------------
// Kernel 4/5: adaptive-Chebyshev GCN, fused.
//   MODE 0 (O=128): zr = sigmoid(gcn(concat[0, de]))   -> z, rg buffers
//   MODE 1 (O=64) : hc = tanh(gcn(concat[0, z*de]));
//                   state = rg*de + (1-rg)*hc; out = state@out_w[t] + out_b[t]
// Block = (bstep, 32-row n-tile), 4 waves.  x[b] streamed in 64-row chunks
// through LDS; xg accumulated in WMMA registers across K=512; W modulated
// per support into LDS; second GEMM over K=3x68 (zero padded 65->68).
// ---------------------------------------------------------------------------
template <int O, int MODE>
__global__ __launch_bounds__(128) void k_gcn(
    const float* __restrict__ Aws, const float* __restrict__ S2ws,
    const float* __restrict__ de, const float* __restrict__ zin,
    float* __restrict__ zout, float* __restrict__ rgws,
    const float* __restrict__ ne2, const float* __restrict__ wp,
    const float* __restrict__ w0, const float* __restrict__ bp,
    const float* __restrict__ b0, const float* __restrict__ outw,
    const float* __restrict__ outb, float* __restrict__ dout) {
  constexpr int NJ = O / 32;  // output tile jobs per wave
  const int tid = threadIdx.x, lane = tid & 31, wid = tid >> 5;
  const int n0 = blockIdx.x * 32, bstep = blockIdx.y;

  __shared__ float xch[64][68];    // x chunk, col 0 zero, cols 65..67 pad
  __shared__ float sch[2][32][64]; // support chunks (A, S2)
  __shared__ float xg[3][32][68];  // aggregated features per support
  __shared__ float wbuf[68][O];    // modulated weights for current support
  __shared__ float tembS[32];
  __shared__ float bvec[O];

  if (tid < 32) tembS[tid] = ne2[bstep * 32 + tid];
  __syncthreads();
  if (tid < O) {
    float s = b0[tid];
#pragma unroll
    for (int d = 0; d < 32; ++d) s += tembS[d] * bp[d * O + tid];
    bvec[tid] = s;
  }

  const int s_id = wid >> 1, mt = wid & 1;
  const v8f zf = {0.f, 0.f, 0.f, 0.f, 0.f, 0.f, 0.f, 0.f};
  v8f acc[5];
#pragma unroll
  for (int it = 0; it < 5; ++it) acc[it] = zf;

  for (int mc = 0; mc < 512; mc += 64) {
    if (mc + 64 < 512)
      __builtin_prefetch(&de[(bstep * 512 + mc + 64) * 64], 0, 1);
    for (int idx = tid; idx < 64 * 68; idx += 128) {
      const int rr = idx / 68, cc = idx % 68;
      float v = 0.f;
      if (cc >= 1 && cc < 65) {
        const int gi = (bstep * 512 + mc + rr) * 64 + (cc - 1);
        v = de[gi];
        if (MODE == 1) v *= zin[gi];
      }
      xch[rr][cc] = v;
    }
    for (int idx = tid; idx < 2 * 32 * 64; idx += 128) {
      const int ss = idx >> 11, rr = (idx >> 6) & 31, cc = idx & 63;
      const float* Sp = ss ? S2ws : Aws;
      sch[ss][rr][cc] = Sp[(n0 + rr) * 512 + mc + cc];
    }
    __syncthreads();
    if ((n0 & ~63) == mc) {  // identity support: copy own rows
      const int roff = n0 - mc;
      for (int idx = tid; idx < 32 * 68; idx += 128)
        xg[0][idx / 68][idx % 68] = xch[roff + idx / 68][idx % 68];
    }
#pragma unroll 4
    for (int kk = 0; kk < 64; kk += 4) {
      v2f a = fragA(&sch[s_id][0][0], 64, 16 * mt, kk, lane);
#pragma unroll
      for (int it = 0; it < 5; ++it) {
        v2f bb = fragB(&xch[0][0], 68, kk, 16 * it, lane);
        acc[it] = wmma4(a, bb, acc[it]);
      }
    }
    __syncthreads();
  }

  {  // spill xg accumulators to LDS
    const int hl = lane >> 4, r = lane & 15;
#pragma unroll
    for (int it = 0; it < 5; ++it)
#pragma unroll
      for (int rv = 0; rv < 8; ++rv)
        xg[1 + s_id][16 * mt + rv + 8 * hl][16 * it + r] = acc[it][rv];
  }
  __syncthreads();

  v8f acc2[NJ];
#pragma unroll
  for (int jj = 0; jj < NJ; ++jj) acc2[jj] = zf;

  for (int k = 0; k < 3; ++k) {
    // W[k] = temb . wp[:,k] + w[k], zero-padded rows 65..67
    for (int idx = tid; idx < 68 * O; idx += 128) {
      const int ii = idx / O, oo = idx % O;
      float v = 0.f;
      if (ii < 65) {
        v = w0[(k * 65 + ii) * O + oo];
#pragma unroll
        for (int d = 0; d < 32; ++d)
          v += tembS[d] * wp[((d * 3 + k) * 65 + ii) * O + oo];
      }
      wbuf[ii][oo] = v;
    }
    __syncthreads();
    for (int jj = 0; jj < NJ; ++jj) {
      const int j = wid + 4 * jj, jmt = j & 1, jot = j >> 1;
      v8f c = acc2[jj];
#pragma unroll 4
      for (int kk = 0; kk < 68; kk += 4) {
        v2f a = fragA(&xg[k][0][0], 68, 16 * jmt, kk, lane);
        v2f bb = fragB(&wbuf[0][0], O, kk, 16 * jot, lane);
        c = wmma4(a, bb, c);
      }
      acc2[jj] = c;
    }
    __syncthreads();
  }

  const int hl = lane >> 4, r = lane & 15;
  if (MODE == 0) {
    for (int jj = 0; jj < NJ; ++jj) {
      const int j = wid + 4 * jj, jmt = j & 1, jot = j >> 1;
#pragma unroll
      for (int rv = 0; rv < 8; ++rv) {
        const int nn = n0 + 16 * jmt + rv + 8 * hl;
        const int oo = 16 * jot + r;
        const float v = acc2[jj][rv] + bvec[oo];
        const float sg = 1.f / (1.f + __expf(-v));
        const int gi = (bstep * 512 + nn) * 64;
        if (oo < 64) zout[gi + oo] = sg;
        else rgws[gi + oo - 64] = sg;
      }
    }
  } else {
    float(*stbuf)[68] = xg[0];  // reuse as state staging
    for (int jj = 0; jj < NJ; ++jj) {
      const int j = wid + 4 * jj, jmt = j & 1, jot = j >> 1;
#pragma unroll
      for (int rv = 0; rv < 8; ++rv) {
        const int nn = n0 + 16 * jmt + rv + 8 * hl;
        const int oo = 16 * jot + r;
        const float hc = tanhf(acc2[jj][rv] + bvec[oo]);
        const int gi = (bstep * 512 + nn) * 64 + oo;
        const float rg = rgws[gi], dv = de[gi];
        stbuf[16 * jmt + rv + 8 * hl][oo] = rg * dv + (1.f - rg) * hc;
      }
    }
    __syncthreads();
    if (tid < 32) {  // fused horizon projection: out = state @ out_w[t] + out_b[t]
      const int t = bstep % 12;
      float s = outb[t];
#pragma unroll 8
      for (int h = 0; h < 64; ++h) s += stbuf[tid][h] * outw[t * 64 + h];
      dout[bstep * 512 + n0 + tid] = s;
    }
  }
}

// ---------------------------------------------------------------------------
extern "C" void kernel_launch(void* const* d_in, const int* in_sizes, int n_in,
                              void* d_out, int out_size, void* d_ws,
                              size_t ws_size, hipStream_t stream) {
  (void)in_sizes; (void)n_in; (void)out_size; (void)ws_size;
  const float* hn  = (const float*)d_in[2];   // h_n (1,B,N,H)
  const float* ne1 = (const float*)d_in[3];   // node_embedding1 (B,T_IN,TD)
  const float* ne2 = (const float*)d_in[4];   // node_embedding2 (B,HORIZON,TD)
  const float* E   = (const float*)d_in[5];   // node_embeddings (N,ED)
  const float* Wq  = (const float*)d_in[6];
  const float* bq  = (const float*)d_in[7];
  const float* Wk  = (const float*)d_in[8];
  const float* bk  = (const float*)d_in[9];
  const float* Wv  = (const float*)d_in[10];
  const float* bv  = (const float*)d_in[11];
  const float* taw = (const float*)d_in[12];
  const float* tab = (const float*)d_in[13];
  const float* gwp = (const float*)d_in[14];
  const float* gw  = (const float*)d_in[15];
  const float* gbp = (const float*)d_in[16];
  const float* gb  = (const float*)d_in[17];
  const float* uwp = (const float*)d_in[18];
  const float* uw  = (const float*)d_in[19];
  const float* ubp = (const float*)d_in[20];
  const float* ub  = (const float*)d_in[21];
  const float* outw = (const float*)d_in[22];
  const float* outb = (const float*)d_in[23];
  float* out = (float*)d_out;

  float* Aws = (float*)d_ws;                  // 512*512
  float* S2  = Aws + 512 * 512;               // 512*512
  float* de  = S2 + 512 * 512;                // 768*512*64
  float* z   = de + 768 * 512 * 64;           // 768*512*64
  float* rg  = z + 768 * 512 * 64;            // 768*512*64

  k_adj<<<512, 256, 0, stream>>>(E, Aws);
  k_cheb<<<dim3(32, 8), 128, 0, stream>>>(Aws, S2);
  k_attn<<<dim3(16, 64), 128, 0, stream>>>(hn, ne1, ne2, Wq, bq, Wk, bk, Wv,
                                           bv, taw, tab, de);
  k_gcn<128, 0><<<dim3(16, 768), 128, 0, stream>>>(
      Aws, S2, de, de, z, rg, ne2, gwp, gw, gbp, gb, outw, outb, out);
  k_gcn<64, 1><<<dim3(16, 768), 128, 0, stream>>>(
      Aws, S2, de, z, z, rg, ne2, uwp, uw, ubp, ub, outw, outb, out);
}